// Encoder_36404142801038
// MI455X (gfx1250) — compile-verified
//
#include <hip/hip_runtime.h>

// ---------------- problem constants ----------------
#define CIN     128
#define C_      256
#define DSTATE  16
#define DCONV   4
#define DTRANK  16
#define G_      32
#define L_      2048
#define N_      (G_ * L_)     // 65536
#define E_      131072
#define EPS_    1e-5f

typedef float v2f __attribute__((ext_vector_type(2)));
typedef float v8f __attribute__((ext_vector_type(8)));

// ---- CDNA5 async global->LDS staging (GLOBAL_LOAD_ASYNC_TO_LDS_B128) ----
#if defined(__has_builtin)
#  if __has_builtin(__builtin_amdgcn_global_load_async_to_lds_b128)
#    define HAS_ASYNC_LDS 1
#  endif
#endif
#ifndef HAS_ASYNC_LDS
#  define HAS_ASYNC_LDS 0
#endif

typedef int v4i_ __attribute__((vector_size(16)));
typedef __attribute__((address_space(1))) v4i_* gbl4_t;   // global int4*
typedef __attribute__((address_space(3))) v4i_* lds4_t;   // LDS int4*

__device__ __forceinline__ void async_copy16(const float* g, float* l) {
#if HAS_ASYNC_LDS
    __builtin_amdgcn_global_load_async_to_lds_b128((gbl4_t)(g), (lds4_t)(l), 0, 0);
#else
    *(float4*)l = *(const float4*)g;
#endif
}

__device__ __forceinline__ void wait_async_lds() {
#if HAS_ASYNC_LDS
#  if __has_builtin(__builtin_amdgcn_s_wait_asynccnt)
    __builtin_amdgcn_s_wait_asynccnt(0);
#  else
    asm volatile("s_wait_asynccnt 0x0" ::: "memory");
#  endif
#endif
}

// =====================================================================
// WMMA fp32 GEMM:  C[M,N] = act(A[M,K] @ B(+bias)) (+resid)
//   BT=true  -> B element [k][n] = Bsrc[n*ldb + k]   (i.e. X @ W^T)
//   ACT: 0 none, 1 relu, 2 softplus
// Block = 128 threads = 4 waves; block tile 64x64; K chunk 32.
// Wave w computes rows [w*16, w*16+16) x 64 cols -> 4 wmma accumulators.
// LDS layout: As[row][k], Bs[col][k]; K-pairs contiguous + 8B aligned so
// each fragment is a single ds_load_b64 (row stride 36 floats = 144B:
// 16B-aligned for async b128 and a conflict-free bank stride).
// =====================================================================
#define TM 64
#define TN 64
#define TK 32
#define TS (TK + 4)   // 36-float row stride

template<int ACT, bool BT>
__global__ __launch_bounds__(128)
void gemm_wmma_f32(const float* __restrict__ A, const float* __restrict__ B,
                   const float* __restrict__ bias, const float* __restrict__ resid,
                   float* __restrict__ Cout,
                   int M, int Nn, int K, int lda, int ldb, int ldc)
{
    __shared__ float As[TM][TS];
    __shared__ float Bs[TN][TS];

    const int tid  = threadIdx.x;
    const int wave = tid >> 5;
    const int lane = tid & 31;
    const int bm   = blockIdx.x * TM;
    const int bn   = blockIdx.y * TN;

    v8f acc[4];
    #pragma unroll
    for (int j = 0; j < 4; ++j) acc[j] = (v8f){0.f,0.f,0.f,0.f,0.f,0.f,0.f,0.f};

    const int mloc = wave * 16 + (lane & 15);   // local row in As
    const int kh   = (lane >> 4) * 2;           // K sub-offset per half-wave
    const int nsub = (lane & 15);
    const bool fullN = (bn + TN <= Nn);

    for (int k0 = 0; k0 < K; k0 += TK) {
        const bool fullK = (k0 + TK <= K);

        // prefetch next K-chunk of A into cache hierarchy
        if (k0 + TK < K)
            __builtin_prefetch(A + (size_t)(bm + (tid & 63)) * lda + (k0 + TK), 0, 1);

        // ---- stage A tile (64 rows x 32 k) ----
        if (fullK) {
            #pragma unroll
            for (int q = tid; q < TM * (TK / 4); q += 128) {
                int r = q >> 3, c4 = (q & 7) << 2;
                async_copy16(A + (size_t)(bm + r) * lda + (k0 + c4), &As[r][c4]);
            }
        } else {
            for (int t = tid; t < TM * TK; t += 128) {
                int r = t / TK, c = t % TK;
                int gk = k0 + c;
                As[r][c] = (gk < K) ? A[(size_t)(bm + r) * lda + gk] : 0.f;
            }
        }

        // ---- stage B tile into Bs[col][k] ----
        if (BT) {
            if (fullK && fullN) {
                #pragma unroll
                for (int q = tid; q < TN * (TK / 4); q += 128) {
                    int r = q >> 3, c4 = (q & 7) << 2;
                    async_copy16(B + (size_t)(bn + r) * ldb + (k0 + c4), &Bs[r][c4]);
                }
            } else {
                for (int t = tid; t < TN * TK; t += 128) {
                    int n = t / TK, c = t % TK;
                    int gk = k0 + c, gn = bn + n;
                    Bs[n][c] = (gk < K && gn < Nn) ? B[(size_t)gn * ldb + gk] : 0.f;
                }
            }
        } else {
            if (fullK && fullN) {
                for (int q = tid; q < TK * (TN / 4); q += 128) {
                    int r = q >> 4, n4 = (q & 15) << 2;
                    float4 v = *(const float4*)(B + (size_t)(k0 + r) * ldb + (bn + n4));
                    Bs[n4 + 0][r] = v.x; Bs[n4 + 1][r] = v.y;
                    Bs[n4 + 2][r] = v.z; Bs[n4 + 3][r] = v.w;
                }
            } else {
                for (int t = tid; t < TN * TK; t += 128) {
                    int n = t / TK, c = t % TK;
                    int gk = k0 + c, gn = bn + n;
                    Bs[n][c] = (gk < K && gn < Nn) ? B[(size_t)gk * ldb + gn] : 0.f;
                }
            }
        }

        if (fullK) wait_async_lds();
        __syncthreads();

        #pragma unroll
        for (int kk = 0; kk < TK; kk += 4) {
            v2f a = *(const v2f*)&As[mloc][kk + kh];
            #pragma unroll
            for (int j = 0; j < 4; ++j) {
                v2f b = *(const v2f*)&Bs[j * 16 + nsub][kk + kh];
                acc[j] = __builtin_amdgcn_wmma_f32_16x16x4_f32(
                    false, a, false, b, (short)0, acc[j], false, false);
            }
        }
        __syncthreads();
    }

    // ---- epilogue: bias, activation, residual, store ----
    const int mbase = bm + wave * 16 + (lane >> 4) * 8;
    #pragma unroll
    for (int j = 0; j < 4; ++j) {
        int gn = bn + j * 16 + nsub;
        if (gn >= Nn) continue;
        float bv = bias ? bias[gn] : 0.f;
        #pragma unroll
        for (int r = 0; r < 8; ++r) {
            int gm = mbase + r;
            if (gm >= M) continue;
            float v = acc[j][r] + bv;
            if (ACT == 1) v = fmaxf(v, 0.f);
            else if (ACT == 2) v = (v > 20.f) ? v : log1pf(__expf(v));
            if (resid) v += resid[(size_t)gm * ldc + gn];
            Cout[(size_t)gm * ldc + gn] = v;
        }
    }
}

// =====================================================================
// GCN helper kernels
// =====================================================================
__global__ void deg_init_kernel(float* deg) {
    int i = blockIdx.x * blockDim.x + threadIdx.x;
    if (i < N_) deg[i] = 1.f;   // self loop
}

__global__ void deg_edges_kernel(const int* __restrict__ ei, float* deg) {
    int e = blockIdx.x * blockDim.x + threadIdx.x;
    if (e < E_) atomicAdd(&deg[ei[E_ + e]], 1.f);
}

__global__ void dis_kernel(const float* __restrict__ deg, float* dis) {
    int i = blockIdx.x * blockDim.x + threadIdx.x;
    if (i < N_) dis[i] = rsqrtf(fmaxf(deg[i], 1.f));
}

// agg[i,c] = b_gcn[c] + XW[i,c] * dis[i]^2   (self-loop term + bias)
__global__ void agg_init_kernel(const float* __restrict__ XW, const float* __restrict__ dis,
                                const float* __restrict__ b_gcn, float* agg) {
    size_t t = (size_t)blockIdx.x * blockDim.x + threadIdx.x;
    size_t total = (size_t)N_ * C_;
    for (; t < total; t += (size_t)gridDim.x * blockDim.x) {
        int c = (int)(t % C_);
        size_t i = t / C_;
        float d = dis[i];
        agg[t] = b_gcn[c] + XW[t] * d * d;
    }
}

__global__ void gcn_scatter_kernel(const int* __restrict__ ei, const float* __restrict__ XW,
                                   const float* __restrict__ dis, float* agg) {
    size_t t = (size_t)blockIdx.x * blockDim.x + threadIdx.x;
    size_t total = (size_t)E_ * C_;
    for (; t < total; t += (size_t)gridDim.x * blockDim.x) {
        int e = (int)(t / C_);
        int c = (int)(t % C_);
        int s = ei[e];
        int d = ei[E_ + e];
        float w = dis[s] * dis[d];
        atomicAdd(&agg[(size_t)d * C_ + c], XW[(size_t)s * C_ + c] * w);
    }
}

// =====================================================================
// BatchNorm (feature-wise over N): stats then apply
// stats layout: [0..C) mean, [C..2C) rstd
// =====================================================================
__global__ __launch_bounds__(256)
void bn_stats_kernel(const float* __restrict__ a, const float* __restrict__ b,
                     float* __restrict__ stats, int ld) {
    int c = blockIdx.x;
    float s = 0.f, s2 = 0.f;
    for (int i = threadIdx.x; i < N_; i += 256) {
        float v = a[(size_t)i * ld + c];
        if (b) v += b[(size_t)i * ld + c];
        s += v; s2 += v * v;
    }
    __shared__ float rs[256], rs2[256];
    rs[threadIdx.x] = s; rs2[threadIdx.x] = s2;
    __syncthreads();
    for (int o = 128; o > 0; o >>= 1) {
        if (threadIdx.x < o) {
            rs[threadIdx.x]  += rs[threadIdx.x + o];
            rs2[threadIdx.x] += rs2[threadIdx.x + o];
        }
        __syncthreads();
    }
    if (threadIdx.x == 0) {
        float mean = rs[0] / (float)N_;
        float var  = rs2[0] / (float)N_ - mean * mean;
        stats[c]      = mean;
        stats[C_ + c] = rsqrtf(var + EPS_);
    }
}

// out = bn(a (+b)) * gamma + beta  (+addp) (relu?)
__global__ void bn_apply_kernel(const float* __restrict__ a, const float* __restrict__ b,
                                const float* __restrict__ addp,
                                const float* __restrict__ stats,
                                const float* __restrict__ gamma, const float* __restrict__ beta,
                                float* __restrict__ out, int relu) {
    size_t t = (size_t)blockIdx.x * blockDim.x + threadIdx.x;
    size_t total = (size_t)N_ * C_;
    for (; t < total; t += (size_t)gridDim.x * blockDim.x) {
        int c = (int)(t % C_);
        float v = a[t];
        if (b) v += b[t];
        v = (v - stats[c]) * stats[C_ + c] * gamma[c] + beta[c];
        if (addp) v += addp[t];
        if (relu) v = fmaxf(v, 0.f);
        out[t] = v;
    }
}

// =====================================================================
// Mamba: causal depth-conv (DCONV=4) + SiLU; x taken from first half of xz
// =====================================================================
__global__ void conv_silu_kernel(const float* __restrict__ xz,
                                 const float* __restrict__ conv_w,
                                 const float* __restrict__ conv_b,
                                 float* __restrict__ xconv) {
    size_t t = (size_t)blockIdx.x * blockDim.x + threadIdx.x;
    size_t total = (size_t)N_ * C_;
    for (; t < total; t += (size_t)gridDim.x * blockDim.x) {
        int c = (int)(t % C_);
        size_t i = t / C_;
        int l = (int)(i % L_);
        float acc = conv_b[c];
        #pragma unroll
        for (int kk = 0; kk < DCONV; ++kk) {
            int lq = l - (DCONV - 1) + kk;
            if (lq >= 0)
                acc += xz[(i - (DCONV - 1) + kk) * (2 * C_) + c] * conv_w[c * DCONV + kk];
        }
        acc = acc / (1.f + __expf(-acc));  // silu
        xconv[t] = acc;
    }
}

// =====================================================================
// Mamba selective scan: 1 block per graph, 1 thread per channel.
// State h[DSTATE] in registers; B/C vectors staged through LDS each step.
// y = (scan_y + x*Dp) * silu(z)
// =====================================================================
__global__ __launch_bounds__(256)
void mamba_scan_kernel(const float* __restrict__ xconv, const float* __restrict__ xz,
                       const float* __restrict__ xdbl, const float* __restrict__ dt,
                       const float* __restrict__ A_log, const float* __restrict__ Dp,
                       float* __restrict__ yout) {
    const int g = blockIdx.x;
    const int c = threadIdx.x;
    __shared__ float sB[DSTATE], sC[DSTATE];

    float A[DSTATE];
    #pragma unroll
    for (int n = 0; n < DSTATE; ++n) A[n] = -__expf(A_log[c * DSTATE + n]);
    float h[DSTATE];
    #pragma unroll
    for (int n = 0; n < DSTATE; ++n) h[n] = 0.f;
    const float dp = Dp[c];

    for (int l = 0; l < L_; ++l) {
        size_t i = (size_t)g * L_ + l;
        __syncthreads();  // protect previous-step reads of sB/sC
        if (c < DSTATE)            sB[c]          = xdbl[i * 48 + DTRANK + c];
        else if (c < 2 * DSTATE)   sC[c - DSTATE] = xdbl[i * 48 + DTRANK + DSTATE + (c - DSTATE)];
        __syncthreads();

        float dtv = dt[i * C_ + c];
        float xv  = xconv[i * C_ + c];
        float zv  = xz[i * (2 * C_) + C_ + c];
        float dx  = dtv * xv;
        float y   = 0.f;
        #pragma unroll
        for (int n = 0; n < DSTATE; ++n) {
            h[n] = __expf(dtv * A[n]) * h[n] + dx * sB[n];
            y += h[n] * sC[n];
        }
        float sz = zv / (1.f + __expf(-zv));
        yout[i * C_ + c] = (y + xv * dp) * sz;
    }
}

// =====================================================================
// Host side
// =====================================================================
static inline float* carve(char*& p, size_t nfloats) {
    float* r = (float*)p;
    p += ((nfloats * sizeof(float) + 255) & ~(size_t)255);
    return r;
}

extern "C" void kernel_launch(void* const* d_in, const int* in_sizes, int n_in,
                              void* d_out, int out_size, void* d_ws, size_t ws_size,
                              hipStream_t stream) {
    const float* nf       = (const float*)d_in[0];
    const int*   ei       = (const int*)  d_in[1];
    // d_in[2] = batch (unused: layout is a pure reshape)
    const float* W_in     = (const float*)d_in[3];
    const float* b_in     = (const float*)d_in[4];
    const float* W_gcn    = (const float*)d_in[5];
    const float* b_gcn    = (const float*)d_in[6];
    const float* gamma1   = (const float*)d_in[7];
    const float* beta1    = (const float*)d_in[8];
    const float* gamma2   = (const float*)d_in[9];
    const float* beta2    = (const float*)d_in[10];
    const float* gamma3   = (const float*)d_in[11];
    const float* beta3    = (const float*)d_in[12];
    const float* W_inproj = (const float*)d_in[13];
    const float* conv_w   = (const float*)d_in[14];
    const float* conv_b   = (const float*)d_in[15];
    const float* W_xproj  = (const float*)d_in[16];
    const float* W_dt     = (const float*)d_in[17];
    const float* b_dt     = (const float*)d_in[18];
    const float* A_log    = (const float*)d_in[19];
    const float* Dp       = (const float*)d_in[20];
    const float* W_outprj = (const float*)d_in[21];
    const float* W_mlp1   = (const float*)d_in[22];
    const float* b_mlp1   = (const float*)d_in[23];
    const float* W_mlp2   = (const float*)d_in[24];
    const float* b_mlp2   = (const float*)d_in[25];
    float* out = (float*)d_out;

    // ---- workspace carve (with reuse) ----
    char* p = (char*)d_ws;
    float* X1    = carve(p, (size_t)N_ * C_);       // x = nf@W_in + b_in
    float* XW    = carve(p, (size_t)N_ * C_);       // x@W_gcn ; later: mamba out, mlp out
    float* AGG   = carve(p, (size_t)N_ * C_);       // gcn agg ; later: scan y, fused out
    float* H1    = carve(p, (size_t)N_ * C_);       // bn1 result
    float* XCV   = carve(p, (size_t)N_ * C_);       // conv+silu
    float* DT    = carve(p, (size_t)N_ * C_);       // softplus(dt)
    float* XD    = carve(p, (size_t)N_ * 48);       // x_dbl
    float* XZ    = carve(p, (size_t)N_ * 2 * C_);   // in-proj ; later: mlp hidden
    float* DEG   = carve(p, N_);
    float* DIS   = carve(p, N_);
    float* ST1   = carve(p, 2 * C_);
    float* ST2   = carve(p, 2 * C_);
    float* ST3   = carve(p, 2 * C_);
    (void)ws_size; (void)n_in; (void)in_sizes; (void)out_size;

    const dim3 gblk(128);
    const int GRID_M = N_ / TM;   // 1024
    const dim3 ew(256);
    const int EWG = 2048;

    // 1) X1 = nf @ W_in + b_in                       [N,128]@[128,256]
    gemm_wmma_f32<0,false><<<dim3(GRID_M, C_/TN), gblk, 0, stream>>>(
        nf, W_in, b_in, nullptr, X1, N_, C_, CIN, CIN, C_, C_);

    // 2) XW = X1 @ W_gcn                             [N,256]@[256,256]
    gemm_wmma_f32<0,false><<<dim3(GRID_M, C_/TN), gblk, 0, stream>>>(
        X1, W_gcn, nullptr, nullptr, XW, N_, C_, C_, C_, C_, C_);

    // 3) GCN degree / norm / scatter
    deg_init_kernel<<<(N_ + 255)/256, ew, 0, stream>>>(DEG);
    deg_edges_kernel<<<(E_ + 255)/256, ew, 0, stream>>>(ei, DEG);
    dis_kernel<<<(N_ + 255)/256, ew, 0, stream>>>(DEG, DIS);
    agg_init_kernel<<<EWG, ew, 0, stream>>>(XW, DIS, b_gcn, AGG);
    gcn_scatter_kernel<<<EWG, ew, 0, stream>>>(ei, XW, DIS, AGG);

    // 4) h1 = bn1(agg + X1)
    bn_stats_kernel<<<C_, 256, 0, stream>>>(AGG, X1, ST1, C_);
    bn_apply_kernel<<<EWG, ew, 0, stream>>>(AGG, X1, nullptr, ST1, gamma1, beta1, H1, 0);

    // 5) Mamba in-proj: XZ = X1 @ W_inproj^T         [N,256]@[256,512]
    gemm_wmma_f32<0,true><<<dim3(GRID_M, (2*C_)/TN), gblk, 0, stream>>>(
        X1, W_inproj, nullptr, nullptr, XZ, N_, 2*C_, C_, C_, C_, 2*C_);

    // 6) causal conv + silu
    conv_silu_kernel<<<EWG, ew, 0, stream>>>(XZ, conv_w, conv_b, XCV);

    // 7) x_dbl = xconv @ W_xproj^T                   [N,256]@[256,48]
    gemm_wmma_f32<0,true><<<dim3(GRID_M, 1), gblk, 0, stream>>>(
        XCV, W_xproj, nullptr, nullptr, XD, N_, 48, C_, C_, C_, 48);

    // 8) dt = softplus(dt_r @ W_dt^T + b_dt)         [N,16]@[16,256]
    gemm_wmma_f32<2,true><<<dim3(GRID_M, C_/TN), gblk, 0, stream>>>(
        XD, W_dt, b_dt, nullptr, DT, N_, C_, DTRANK, 48, DTRANK, C_);

    // 9) selective scan -> y (reuse AGG)
    mamba_scan_kernel<<<G_, C_, 0, stream>>>(XCV, XZ, XD, DT, A_log, Dp, AGG);

    // 10) hm = y @ W_outproj^T  (reuse XW)           [N,256]@[256,256]
    gemm_wmma_f32<0,true><<<dim3(GRID_M, C_/TN), gblk, 0, stream>>>(
        AGG, W_outprj, nullptr, nullptr, XW, N_, C_, C_, C_, C_, C_);

    // 11) out = h1 + bn2(hm + X1)  (reuse AGG as OUT)
    bn_stats_kernel<<<C_, 256, 0, stream>>>(XW, X1, ST2, C_);
    bn_apply_kernel<<<EWG, ew, 0, stream>>>(XW, X1, H1, ST2, gamma2, beta2, AGG, 0);

    // 12) mlp hidden = relu(out @ W_mlp1 + b_mlp1)   [N,256]@[256,512] (reuse XZ)
    gemm_wmma_f32<1,false><<<dim3(GRID_M, (2*C_)/TN), gblk, 0, stream>>>(
        AGG, W_mlp1, b_mlp1, nullptr, XZ, N_, 2*C_, C_, C_, 2*C_, 2*C_);

    // 13) s3 = out + hidden @ W_mlp2 + b_mlp2        [N,512]@[512,256] (reuse XW)
    gemm_wmma_f32<0,false><<<dim3(GRID_M, C_/TN), gblk, 0, stream>>>(
        XZ, W_mlp2, b_mlp2, AGG, XW, N_, C_, 2*C_, 2*C_, C_, C_);

    // 14) out = relu(bn3(s3))
    bn_stats_kernel<<<C_, 256, 0, stream>>>(XW, nullptr, ST3, C_);
    bn_apply_kernel<<<EWG, ew, 0, stream>>>(XW, nullptr, nullptr, ST3, gamma3, beta3, out, 1);
}